// KPConvLayer_80401787781839
// MI455X (gfx1250) — compile-verified
//
#include <hip/hip_runtime.h>

typedef __attribute__((ext_vector_type(2))) float v2f;
typedef __attribute__((ext_vector_type(4))) float v4f;
typedef __attribute__((ext_vector_type(8))) float v8f;

#define POINT_INFLUENCE 0.06f
#define N_Q   32768
#define N_SUP 40000
#define M_NEI 32
#define K_PTS 15
#define C_IN  64
#define C_OUT 128
#define KC    (K_PTS * C_IN)      // 960 = GEMM contraction length
#define QB    32                  // queries per block (2 row-tiles of 16)
#define WSTRIDE 968               // 960 + 8 pad (stride mod 64 banks = 8)

// LDS layout (floats):
//   w_lds   : [QB][32][16]        = 16384
//   wted    : [QB][WSTRIDE]       = 30976
//   idx_lds : [QB][32] (int)      = 1024
//   kp_lds  : [15][3] (+pad)      = 48
#define SM_WLDS  0
#define SM_WTED  16384
#define SM_IDX   (SM_WTED + QB * WSTRIDE)
#define SM_KP    (SM_IDX + QB * M_NEI)
#define SM_TOTAL (SM_KP + 48)

__global__ __launch_bounds__(256) void kpconv_fused_kernel(
    const float* __restrict__ qpts,        // [N,3]
    const float* __restrict__ spts,        // [N0,3]
    const long long* __restrict__ nbrs,    // [N,M] int64
    const float* __restrict__ feats,       // [N0,Cin]
    const float* __restrict__ kpos,        // [K,3]
    const float* __restrict__ wgt,         // [K*Cin, Cout] (k*Cin+c major)
    float* __restrict__ out)               // [N, Cout]
{
  extern __shared__ float smem[];
  float* w_lds   = smem + SM_WLDS;
  float* wted    = smem + SM_WTED;
  int*   idx_lds = (int*)(smem + SM_IDX);
  float* kp_lds  = smem + SM_KP;

  const int t  = threadIdx.x;
  const int n0 = blockIdx.x * QB;

  // ---- stage kernel-point positions -------------------------------------
  if (t < K_PTS * 3) kp_lds[t] = kpos[t];
  __syncthreads();

  // ---- Phase 1: per (query, neighbor) kernel-point influences -----------
  #pragma unroll
  for (int p = t; p < QB * M_NEI; p += 256) {
    const int n = p >> 5;
    const int m = p & 31;
    long long idx = nbrs[(long long)(n0 + n) * M_NEI + m];
    int ii = (int)idx;
    float px, py, pz;
    if (ii < N_SUP) {
      px = spts[ii * 3 + 0]; py = spts[ii * 3 + 1]; pz = spts[ii * 3 + 2];
    } else {
      px = py = pz = 1.0e6f;   // shadow point -> influence clamps to 0
      ii = 0;                  // safe gather index; weight is exactly 0
    }
    const float qx = qpts[(n0 + n) * 3 + 0];
    const float qy = qpts[(n0 + n) * 3 + 1];
    const float qz = qpts[(n0 + n) * 3 + 2];
    const float dx = px - qx, dy = py - qy, dz = pz - qz;
    idx_lds[n * M_NEI + m] = ii;
    #pragma unroll
    for (int k = 0; k < 16; ++k) {
      float w = 0.0f;
      if (k < K_PTS) {
        const float ex = dx - kp_lds[k * 3 + 0];
        const float ey = dy - kp_lds[k * 3 + 1];
        const float ez = dz - kp_lds[k * 3 + 2];
        const float sq = ex * ex + ey * ey + ez * ez;
        w = fmaxf(0.0f, 1.0f - sqrtf(sq) * (1.0f / POINT_INFLUENCE));
      }
      w_lds[(n * M_NEI + m) * 16 + k] = w;
    }
  }
  __syncthreads();

  // ---- Phase 2+3: normalize over m, aggregate neighbor features ---------
  // Slot s = (n, k), k fastest: 16 lanes share one feature row (broadcast).
  #pragma unroll
  for (int half = 0; half < 2; ++half) {
    const int k = t & 15;
    const int n = (t >> 4) + half * 16;
    if (k < K_PTS) {
      float s = 0.0f;
      #pragma unroll
      for (int m = 0; m < M_NEI; ++m) s += w_lds[(n * M_NEI + m) * 16 + k];
      const float inv = 1.0f / (s + 1e-10f);

      v4f acc[16];
      #pragma unroll
      for (int i = 0; i < 16; ++i) { v4f z = {0.f, 0.f, 0.f, 0.f}; acc[i] = z; }

      for (int m = 0; m < M_NEI; ++m) {
        const float wv = w_lds[(n * M_NEI + m) * 16 + k] * inv;
        const v4f* frow = (const v4f*)(feats + (long long)idx_lds[n * M_NEI + m] * C_IN);
        #pragma unroll
        for (int i = 0; i < 16; ++i) acc[i] += frow[i] * wv;
      }
      v4f* dst = (v4f*)(wted + n * WSTRIDE + k * C_IN);
      #pragma unroll
      for (int i = 0; i < 16; ++i) dst[i] = acc[i];
    }
  }
  __syncthreads();

  // ---- Stage B: out[32x128] = wted[32x960] @ wgt[960x128] via f32 WMMA --
  // Each wave owns one 16-wide Cout tile and BOTH 16-row query tiles, so
  // every B fragment loaded from L2 feeds two WMMAs.
  {
    const int wave = t >> 5;               // 0..7 -> Cout tile
    const int lane = t & 31;
    const int d0   = wave * 16;
    const int row  = lane & 15;            // M for A, N for B
    const int koff = (lane >> 4) * 2;      // VGPR pair carries K {koff, koff+1}
    const float* arow0 = wted + row * WSTRIDE + koff;            // rows 0..15
    const float* arow1 = wted + (row + 16) * WSTRIDE + koff;     // rows 16..31
    const float* bcol  = wgt + (long long)koff * C_OUT + d0 + row;

    v8f c0 = {};
    v8f c1 = {};
    #pragma unroll 4
    for (int kk = 0; kk < KC; kk += 4) {
      // keep the B stream flowing into near caches (global_prefetch_b8)
      __builtin_prefetch(bcol + (long long)(kk + 32) * C_OUT, 0, 3);
      v2f b;
      b.x = bcol[(long long)kk * C_OUT];                // B[kk+koff][d0+row]
      b.y = bcol[(long long)kk * C_OUT + C_OUT];        // B[kk+koff+1][d0+row]
      v2f a0 = *(const v2f*)(arow0 + kk);               // ds_load_b64
      v2f a1 = *(const v2f*)(arow1 + kk);
      c0 = __builtin_amdgcn_wmma_f32_16x16x4_f32(
              false, a0, false, b, (short)0, c0, false, false);
      c1 = __builtin_amdgcn_wmma_f32_16x16x4_f32(
              false, a1, false, b, (short)0, c1, false, false);
    }

    // C/D layout: VGPR r -> M = r + 8*(lane>>4), N = lane&15
    const int mbase = (lane >> 4) * 8;
    float* ocol0 = out + (long long)n0 * C_OUT + d0 + (lane & 15);
    float* ocol1 = ocol0 + 16LL * C_OUT;
    #pragma unroll
    for (int r = 0; r < 8; ++r) {
      ocol0[(long long)(mbase + r) * C_OUT] = c0[r];
      ocol1[(long long)(mbase + r) * C_OUT] = c1[r];
    }
  }
}

extern "C" void kernel_launch(void* const* d_in, const int* in_sizes, int n_in,
                              void* d_out, int out_size, void* d_ws, size_t ws_size,
                              hipStream_t stream) {
  (void)in_sizes; (void)n_in; (void)out_size; (void)d_ws; (void)ws_size;
  const float*     qpts = (const float*)d_in[0];
  const float*     spts = (const float*)d_in[1];
  const long long* nbrs = (const long long*)d_in[2];
  const float*     x    = (const float*)d_in[3];
  const float*     kpos = (const float*)d_in[4];
  const float*     wgt  = (const float*)d_in[5];
  float*           out  = (float*)d_out;

  const dim3 grid(N_Q / QB);
  const dim3 block(256);
  const size_t shmem = (size_t)SM_TOTAL * sizeof(float);
  hipLaunchKernelGGL(kpconv_fused_kernel, grid, block, shmem, stream,
                     qpts, spts, nbrs, x, kpos, wgt, out);
}